// ROIPooling_48430051230364
// MI455X (gfx1250) — compile-verified
//
#include <hip/hip_runtime.h>
#include <stdint.h>

// Problem constants (from reference setup_inputs)
#define BB 4
#define CC 64
#define HH 128
#define WW 128
#define NN 64
#define BOXD 7
#define PH 7
#define PW 7
#define NWAVES 8
#define CHUNK 10          // rows per TDM chunk (uniform; last chunk may be short)

typedef unsigned int u32x4 __attribute__((ext_vector_type(4)));
typedef int          i32x4 __attribute__((ext_vector_type(4)));
typedef int          i32x8 __attribute__((ext_vector_type(8)));

// Issue a 2-D TDM tile load: tileRows x tileW f32 elements, row stride = WW
// floats, from global 'gaddr' into LDS byte address 'lds_addr' (packed rows).
__device__ __forceinline__ void tdm_load_2d(unsigned lds_addr,
                                            unsigned long long gaddr,
                                            int tileW, int tileRows) {
  // --- D# group 0 (128b): count=1, lds_addr, global_addr(57b), type=2 ---
  u32x4 g0;
  g0[0] = 1u;                                        // count=1 (valid), user mode
  g0[1] = lds_addr;                                  // LDS byte address
  g0[2] = (unsigned)(gaddr & 0xFFFFFFFFull);         // global addr [31:0]
  g0[3] = (unsigned)((gaddr >> 32) & 0x1FFFFFFull)   // global addr [56:32]
          | (2u << 30);                              // type = 2 ("image")
  // --- D# group 1 (256b) ---
  const unsigned TD = 1u << 20;                      // huge tensor dims: never OOB
  i32x8 g1;
  g1[0] = (int)(2u << 16);                           // wg_mask=0, data_size=2 (4B)
  g1[1] = (int)((TD & 0xFFFFu) << 16);               // barrier_addr=0 | tdim0[15:0]
  g1[2] = (int)((TD >> 16) | ((TD & 0xFFFFu) << 16));// tdim0[31:16] | tdim1[15:0]
  g1[3] = (int)((TD >> 16) | ((unsigned)tileW << 16));// tdim1[31:16] | tile_dim0
  g1[4] = (int)((unsigned)tileRows & 0xFFFFu);       // tile_dim1 | tile_dim2=0
  g1[5] = WW;                                        // tensor_dim0_stride = 128
  g1[6] = 0;
  g1[7] = 0;
  i32x4 z4 = {0, 0, 0, 0};
  asm volatile("" ::: "memory");
#if defined(__clang_major__) && __clang_major__ >= 23
  i32x8 z8 = {0, 0, 0, 0, 0, 0, 0, 0};
  __builtin_amdgcn_tensor_load_to_lds(g0, g1, z4, z4, z8, 0);
#else
  __builtin_amdgcn_tensor_load_to_lds(g0, g1, z4, z4, 0);
#endif
}

__device__ __forceinline__ void wait_tensor_le1() {
#if __has_builtin(__builtin_amdgcn_s_wait_tensorcnt)
  __builtin_amdgcn_s_wait_tensorcnt(1);
#else
  asm volatile("s_wait_tensorcnt 0x1");
#endif
  asm volatile("" ::: "memory");
}
__device__ __forceinline__ void wait_tensor_le0() {
#if __has_builtin(__builtin_amdgcn_s_wait_tensorcnt)
  __builtin_amdgcn_s_wait_tensorcnt(0);
#else
  asm volatile("s_wait_tensorcnt 0x0");
#endif
  asm volatile("" ::: "memory");
}

__global__ __launch_bounds__(256, 1)
void ROIPooling_48430051230364_kernel(const float* __restrict__ feat,
                                      const float* __restrict__ boxes,
                                      float* __restrict__ out) {
  __shared__ float tile[NWAVES][2][CHUNK * WW];  // 80 KB: double-buffered DMA tiles
  __shared__ float pybuf[NWAVES][PH][WW];        // 28 KB: row-pooled intermediates

  const int bn   = blockIdx.x;          // bn = b*NN + n
  const int tid  = threadIdx.x;
  const int wave = tid >> 5;
  const int lane = tid & 31;

  const float* box = boxes + (size_t)bn * BOXD;
  bool valid = false;
#pragma unroll
  for (int d = 0; d < BOXD; ++d) valid |= (box[d] != 0.0f);

  float* outbox = out + (size_t)bn * (CC * PH * PW);
  if (!valid) {  // all-zero box -> zero fill (uniform per block)
    for (int i = tid; i < CC * PH * PW; i += 256) outbox[i] = 0.0f;
    return;
  }

  // Faithful replication of the reference coordinate math.
  float x1f = fminf(fmaxf(box[0], 0.0f), (float)(WW - 1));
  float y1f = fminf(fmaxf(box[1], 0.0f), (float)(HH - 1));
  float x2f = fminf(fmaxf(box[2], 0.0f), (float)(WW - 1));
  float y2f = fminf(fmaxf(box[3], 0.0f), (float)(HH - 1));
  x1f = fminf(x1f, x2f);   // x2 = max(new_x1, x2) == x2, so x2f unchanged
  y1f = fminf(y1f, y2f);
  int x1i = (int)x1f, x2i = (int)x2f, y1i = (int)y1f, y2i = (int)y2f;
  if (x2i == x1i) x2i = x1i + 1;
  if (y2i == y1i) y2i = y1i + 1;
  const int Lh = y2i - y1i;   // [1, 128]
  const int Lw = x2i - x1i;   // [1, 128]

  int ys[PH], ye[PH];
#pragma unroll
  for (int p = 0; p < PH; ++p) {
    ys[p] = y1i + (p * Lh) / PH;                  // floor
    ye[p] = y1i + ((p + 1) * Lh + PH - 1) / PH;   // ceil
  }

  const int b = bn / NN;
  const unsigned long long featBase = (unsigned long long)(uintptr_t)feat;

  const int NCHK   = (Lh + CHUNK - 1) / CHUNK;    // chunks per channel (uniform)
  const int nTiles = (CC / NWAVES) * NCHK;        // flat pipeline length (>= 8)

  // Issue chunk 'idx' (channel-major order) into tile[wave][idx&1].
  auto issueIdx = [&](int idx) {
    const int ci = idx / NCHK;
    const int k  = idx - ci * NCHK;
    const int c  = wave + ci * NWAVES;
    const int rowstart = y1i + k * CHUNK;
    const int rows = min(CHUNK, y2i - rowstart);
    unsigned long long g = featBase +
        4ull * ((((size_t)b * CC + c) * HH + (size_t)rowstart) * (size_t)WW
                + (size_t)x1i);
    tdm_load_2d((unsigned)(uintptr_t)&tile[wave][idx & 1][0], g, Lw, rows);
  };

  float acc[PH][4];
#pragma unroll
  for (int p = 0; p < PH; ++p)
#pragma unroll
    for (int k = 0; k < 4; ++k) acc[p][k] = -INFINITY;

  // Prologue: fill both pipeline slots.
  issueIdx(0);
  issueIdx(1);

  for (int idx = 0; idx < nTiles; ++idx) {
    // In-order per-wave TDM completion: <=1 outstanding => oldest (idx) done.
    if (nTiles - idx >= 2) wait_tensor_le1();
    else                   wait_tensor_le0();

    const int ci = idx / NCHK;
    const int k  = idx - ci * NCHK;
    const int rowstart = y1i + k * CHUNK;
    const int rows = min(CHUNK, y2i - rowstart);
    const float* tbuf = &tile[wave][idx & 1][0];

    for (int j = 0; j < rows; ++j) {
      const int y = rowstart + j;
      const float* rp = tbuf + j * Lw;             // packed row of width Lw
      float v0 = -INFINITY, v1 = -INFINITY, v2 = -INFINITY, v3 = -INFINITY;
      if (lane      < Lw) v0 = rp[lane];
      if (lane + 32 < Lw) v1 = rp[lane + 32];
      if (lane + 64 < Lw) v2 = rp[lane + 64];
      if (lane + 96 < Lw) v3 = rp[lane + 96];
#pragma unroll
      for (int p = 0; p < PH; ++p) {
        if (y >= ys[p] && y < ye[p]) {             // row may belong to many bins
          acc[p][0] = fmaxf(acc[p][0], v0);
          acc[p][1] = fmaxf(acc[p][1], v1);
          acc[p][2] = fmaxf(acc[p][2], v2);
          acc[p][3] = fmaxf(acc[p][3], v3);
        }
      }
    }

    // Keep the pipeline full: DMA for idx+2 overlaps everything below.
    if (idx + 2 < nTiles) issueIdx(idx + 2);

    if (k == NCHK - 1) {                           // channel complete: flush
      const int c = wave + ci * NWAVES;
#pragma unroll
      for (int p = 0; p < PH; ++p) {
        if (lane      < Lw) pybuf[wave][p][lane]      = acc[p][0];
        if (lane + 32 < Lw) pybuf[wave][p][lane + 32] = acc[p][1];
        if (lane + 64 < Lw) pybuf[wave][p][lane + 64] = acc[p][2];
        if (lane + 96 < Lw) pybuf[wave][p][lane + 96] = acc[p][3];
        acc[p][0] = acc[p][1] = acc[p][2] = acc[p][3] = -INFINITY;
      }
      // Stage 2: column bins (same wave wrote pybuf; DS is in-order per wave).
      float* outc = outbox + (size_t)c * (PH * PW);
      for (int o = lane; o < PH * PW; o += 32) {
        const int p  = o / PW;
        const int q  = o % PW;
        const int xs = (q * Lw) / PW;              // relative to x1i
        const int xe = ((q + 1) * Lw + PW - 1) / PW;
        float m = -INFINITY;
        for (int x = xs; x < xe; ++x) m = fmaxf(m, pybuf[wave][p][x]);
        outc[o] = m;
      }
    }
  }
}

extern "C" void kernel_launch(void* const* d_in, const int* in_sizes, int n_in,
                              void* d_out, int out_size, void* d_ws, size_t ws_size,
                              hipStream_t stream) {
  (void)in_sizes; (void)n_in; (void)d_ws; (void)ws_size; (void)out_size;
  const float* feat  = (const float*)d_in[0];   // spatial_features [4,64,128,128]
  const float* boxes = (const float*)d_in[1];   // boxes [4,64,7]
  float* out = (float*)d_out;                   // [4,64,64,7,7]
  ROIPooling_48430051230364_kernel<<<dim3(BB * NN), dim3(256), 0, stream>>>(feat, boxes, out);
}